// CritiGraph_86492051406961
// MI455X (gfx1250) — compile-verified
//
#include <hip/hip_runtime.h>

// CritiGraph ct_val kernel for MI455X (gfx1250, wave32).
// T=512, S=64, TP=8, H=12, K=4, C=2*K*H+1=97. Output (T,S,C,TP) f32.
//
// Memory-bound on the 101.7MB output stream (floor ~4.4us @ 23.3TB/s).
// Math is XOR/CLZ-based (no matmul -> no WMMA); CDNA5 path used: TDM async
// tensor loads to LDS + s_wait_tensorcnt, wave32 shuffles, b128 stores.

#define TT   512
#define SSZ  64
#define TPD  8
#define HH   12
#define KK   4
#define CC   97
#define SLOC 32                 // s-values per block (grid.y = 2)
#define NPAIR (SLOC * CC)       // 3104 (s,c) pairs per block

typedef __attribute__((ext_vector_type(4))) unsigned int v4u_t;
typedef __attribute__((ext_vector_type(8))) int          v8i_t;
typedef __attribute__((ext_vector_type(4))) int          v4i_t;

// Low 32 bits of a generic pointer to LDS == LDS byte offset (ISA 10.2:
// the LDS offset is the low 32 address bits).
__device__ __forceinline__ unsigned lds_off_u32(const void* p) {
  return (unsigned)(unsigned long long)(size_t)p;
}

// 1-D contiguous TDM load: copies n_dwords int32 from global gptr to LDS at byte offset lds_byte_off.
// D# per CDNA5 ISA ch.8: group0 = {count=1 | lds_addr | global_addr | type=2},
// group1 = {data_size=4B, tensor_dim0=n, tensor_dim1=1, tile_dim0=n,
//           tile_dim1=1, tensor_dim0_stride=n}. Groups 2/3 zero (<=2D).
__device__ __forceinline__ void tdm_load_1d(unsigned lds_byte_off,
                                            const void* gptr,
                                            unsigned n_dwords) {
  unsigned long long ga = (unsigned long long)(size_t)gptr;
  v4u_t g0;
  g0.x = 1u;                                            // count=1, user mode
  g0.y = lds_byte_off;                                  // lds_addr
  g0.z = (unsigned)ga;                                  // global_addr[31:0]
  g0.w = (unsigned)((ga >> 32) & 0x1FFFFFFull)          // global_addr[56:32]
         | 0x80000000u;                                 // type=2 ("image")
  v8i_t g1;
  g1[0] = (int)(2u << 16);                              // data_size=2 -> 4B
  g1[1] = (int)((n_dwords & 0xFFFFu) << 16);            // tensor_dim0 lo16
  g1[2] = (int)(((n_dwords >> 16) & 0xFFFFu)            // tensor_dim0 hi16
                | (1u << 16));                          // tensor_dim1 = 1
  g1[3] = (int)((n_dwords & 0xFFFFu) << 16);            // tile_dim0 = n
  g1[4] = 1;                                            // tile_dim1 = 1
  g1[5] = (int)n_dwords;                                // tensor_dim0_stride
  g1[6] = 0;
  g1[7] = 0;
  v4i_t z4 = {0, 0, 0, 0};
#if __has_include(<hip/amd_detail/amd_gfx1250_TDM.h>)
  v8i_t z8 = {0, 0, 0, 0, 0, 0, 0, 0};
  __builtin_amdgcn_tensor_load_to_lds(g0, g1, z4, z4, z8, 0);   // clang-23 form
#else
  __builtin_amdgcn_tensor_load_to_lds(g0, g1, z4, z4, 0);       // ROCm 7.2 form
#endif
}

// pack sign|abs:  bit31 = sign(v), bits[11:0] = |v|
__device__ __forceinline__ int comb_pos(int r) {
  int a = r < 0 ? -r : r;
  return (r < 0 ? (int)0x80000000 : 0) | a;
}
__device__ __forceinline__ int comb_neg(int r) {     // sign|abs of (-r)
  int a = r < 0 ? -r : r;
  return (r > 0 ? (int)0x80000000 : 0) | a;
}

__global__ __launch_bounds__(256) void critigraph_ctval_kernel(
    const int* __restrict__ sta,    // (T,TP)
    const int* __restrict__ pos,    // (T,S,TP)
    const float* __restrict__ nrm,  // (T,S)
    const int* __restrict__ rm,     // (T,H,K,TP)
    float* __restrict__ out) {      // (T,S,C,TP)
  __shared__ int   s_pos[SLOC * TPD];     // TDM-staged pos slice
  __shared__ int   s_rm[HH * KK * TPD];   // TDM-staged mask row
  __shared__ int   s_sta[TPD];
  __shared__ int   s_cA[CC * TPD];        // candidates, sign|abs packed
  __shared__ int   s_pA[SLOC * TPD];      // positives,  sign|abs packed
  __shared__ float s_base[SLOC * TPD];    // (sum - cos_d)/8
  __shared__ float s_a[SLOC];             // (norm - 32*norm/12)/8
  __shared__ float s_b[SLOC];             // (norm/12)/8

  const int t   = blockIdx.x;
  const int s0  = blockIdx.y * SLOC;
  const int tid = threadIdx.x;

  if (tid < TPD) s_sta[tid] = sta[t * TPD + tid];

  if (tid < 32) {  // wave 0 only: issue async TDM loads, then drain tensorcnt
    tdm_load_1d(lds_off_u32(s_pos),
                pos + ((size_t)t * SSZ + s0) * TPD, SLOC * TPD);
    tdm_load_1d(lds_off_u32(s_rm),
                rm + (size_t)t * HH * KK * TPD, HH * KK * TPD);
    __builtin_amdgcn_s_wait_tensorcnt(0);
  }
  __syncthreads();

  // ---- candidate table: c<48 -> sta^(1<<h)^mask ; c==48 -> sta ; c>48 -> -(...)
  for (int m = tid; m < CC * TPD; m += 256) {
    int c = m >> 3, d = m & 7;
    int sv = s_sta[d];
    int v;
    if (c < 48) {
      int h = c >> 2;                       // i = h*K + k, K=4
      v = comb_pos(sv ^ (1 << h) ^ s_rm[m]);
    } else if (c == 48) {
      v = comb_pos(sv);
    } else {
      int i = c - 49;
      int h = i >> 2;
      v = comb_neg(sv ^ (1 << h) ^ s_rm[i * TPD + d]);
    }
    s_cA[m] = v;
  }

  // ---- per (s,d): packed positive, cos_sta_pos, row sums (width-8 shuffle)
  {
    int item = tid;                        // SLOC*TPD == 256 == blockDim
    int sl = item >> 3, d = item & 7;
    int pc = comb_pos(s_pos[item]);
    s_pA[item] = pc;
    int sc = comb_pos(s_sta[d]);
    int x  = pc ^ sc;
    float n  = nrm[(size_t)t * SSZ + s0 + sl];
    float n12 = n * (1.0f / 12.0f);
    int   e  = 32 - __clz((x & 0x7fffffff) + 1);   // frexp exponent
    float mag = n - (float)e * n12;
    float cosv = __int_as_float(__float_as_int(mag) ^ (x & (int)0x80000000));
    float sum = cosv;                              // reduce over d (8 lanes)
    sum += __shfl_xor(sum, 1, 32);
    sum += __shfl_xor(sum, 2, 32);
    sum += __shfl_xor(sum, 4, 32);
    s_base[item] = (sum - cosv) * 0.125f;
    if (d == 0) {
      s_a[sl] = (n - 32.0f * n12) * 0.125f;        // fold e = 32 - clz and /TP
      s_b[sl] = n12 * 0.125f;
    }
  }
  __syncthreads();

  // ---- main loop: one (s,c) pair per thread-iteration, 8 outputs, 2x b128
  const size_t outrow = ((size_t)t * SSZ + s0) * CC * TPD;
  for (int j = tid; j < NPAIR; j += 256) {
    int sl = j / CC;
    int c  = j - sl * CC;
    float av = s_a[sl], bv = s_b[sl];
    const int*   pa = &s_pA[sl * TPD];
    const int*   ca = &s_cA[c * TPD];
    const float* bb = &s_base[sl * TPD];
    float o[TPD];
#pragma unroll
    for (int d = 0; d < TPD; ++d) {
      int x = pa[d] ^ ca[d];                       // abs-xor + combined sign
      float clzf = (float)__clz((x & 0x7fffffff) + 1);
      float mag  = fmaf(clzf, bv, av);             // (1 - e/12)*norm / 8
      float r = __int_as_float(__float_as_int(mag) ^ (x & (int)0x80000000));
      o[d] = bb[d] + r;
    }
    float4* dst = (float4*)(out + outrow + (size_t)j * TPD);
    dst[0] = make_float4(o[0], o[1], o[2], o[3]);
    dst[1] = make_float4(o[4], o[5], o[6], o[7]);
  }
}

extern "C" void kernel_launch(void* const* d_in, const int* in_sizes, int n_in,
                              void* d_out, int out_size, void* d_ws,
                              size_t ws_size, hipStream_t stream) {
  (void)in_sizes; (void)n_in; (void)out_size; (void)d_ws; (void)ws_size;
  const int*   sta = (const int*)d_in[0];     // sta_loc (T,TP) int32
  const int*   pos = (const int*)d_in[1];     // pos_loc (T,S,TP) int32
  const float* nrm = (const float*)d_in[2];   // eu_norm (T,S) f32
  const int*   rm  = (const int*)d_in[3];     // random_masks (T,H,K,TP) int32
  float* out = (float*)d_out;                 // (T,S,C,TP) f32

  dim3 grid(TT, SSZ / SLOC, 1);               // (512, 2)
  critigraph_ctval_kernel<<<grid, 256, 0, stream>>>(sta, pos, nrm, rm, out);
}